// RecursiveNetClassifier_4990751998297
// MI455X (gfx1250) — compile-verified
//
#include <hip/hip_runtime.h>
#include <hip/hip_bf16.h>

#define DIMV   256
#define SV     16
#define OUTV   10
#define LDSROW 260   // 256 + 4 pad -> A-fragment LDS reads hit distinct banks

typedef __attribute__((ext_vector_type(2))) float v2f;
typedef __attribute__((ext_vector_type(8))) float v8f;

// ---------------------------------------------------------------------------
// Bucket nodes by (level, symbol) so every GEMM tile shares one W[s].
// ---------------------------------------------------------------------------
__global__ void rnc_zero_counts(int* __restrict__ counts, int n) {
    int i = threadIdx.x;
    if (i < n) counts[i] = 0;
}

__global__ void rnc_bucket(const int* __restrict__ x_sym,
                           int* __restrict__ lists,
                           int* __restrict__ counts,
                           int N) {
    int i = blockIdx.x * 256 + threadIdx.x;
    if (i >= N) return;
    int level = 31 - __clz(i + 1);       // perfect-heap level of node i
    int lo    = (1 << level) - 1;        // first node index of this level
    int s     = x_sym[i];
    int pos   = atomicAdd(&counts[level * SV + s], 1);
    lists[(size_t)s * N + lo + pos] = i;
}

// ---------------------------------------------------------------------------
// One level: for each (symbol, 16-node tile) block, compute
//   enc[node] = tanh(mean(children) @ W[s] + b[s])
// as a 16x256x256 fp32 GEMM on the matrix pipe (V_WMMA_F32_16X16X4_F32).
// 128 threads = 4 waves; each wave owns 64 output columns (4 N-tiles).
// ---------------------------------------------------------------------------
__global__ __launch_bounds__(128) void rnc_level_gemm(
    const int*   __restrict__ child_idx,  // N x 2
    const float* __restrict__ W,          // S x 256 x 256 (row-major [d][e])
    const float* __restrict__ bvec,       // S x 256
    const int*   __restrict__ lists,      // S x N
    const int*   __restrict__ counts,     // depth x S
    float*       __restrict__ enc,        // N x 256
    int level, int lo, int N)
{
    __shared__ float meanLds[16 * LDSROW];
    __shared__ int   nodeIds[16];

    const int s   = blockIdx.x;                 // symbol
    const int t   = blockIdx.y;                 // 16-node tile within bucket
    const int cnt = counts[level * SV + s];
    if (t * 16 >= cnt) return;                  // uniform per block

    const int tid = threadIdx.x;

    // ---- gather node ids for this tile ----
    if (tid < 16) {
        int pos = t * 16 + tid;
        nodeIds[tid] = (pos < cnt) ? lists[(size_t)s * N + lo + pos] : -1;
    }
    __syncthreads();

    // ---- compute child means into LDS (A matrix, 16 rows x 256 cols) ----
    {
        int row  = tid >> 3;       // 0..15
        int sub  = tid & 7;        // 8 threads per row
        int node = nodeIds[row];
        int c0 = -1, c1 = -1;
        if (node >= 0) {
            c0 = child_idx[2 * node];
            c1 = child_idx[2 * node + 1];
        }
        int nch = (c0 >= 0) + (c1 >= 0);
        float inv = (nch > 0) ? (1.0f / (float)nch) : 0.0f;
        const float4* p0 = (c0 >= 0) ? (const float4*)(enc + (size_t)c0 * DIMV) : nullptr;
        const float4* p1 = (c1 >= 0) ? (const float4*)(enc + (size_t)c1 * DIMV) : nullptr;
        #pragma unroll
        for (int m = 0; m < 8; ++m) {
            int k4 = sub + m * 8;                      // float4 chunk 0..63
            float ax = 0.f, ay = 0.f, az = 0.f, aw = 0.f;
            if (p0) { float4 v = p0[k4]; ax += v.x; ay += v.y; az += v.z; aw += v.w; }
            if (p1) { float4 v = p1[k4]; ax += v.x; ay += v.y; az += v.z; aw += v.w; }
            float* dst = &meanLds[row * LDSROW + k4 * 4];
            dst[0] = ax * inv; dst[1] = ay * inv; dst[2] = az * inv; dst[3] = aw * inv;
        }
    }
    __syncthreads();

    // ---- GEMM: A(16x256, LDS) x W[s](256x256) -> C(16x256), fp32 WMMA ----
    const int wave  = tid >> 5;          // 0..3
    const int lane  = tid & 31;
    const int row16 = lane & 15;         // M for A / N-sublane for B,C
    const int khalf = (lane >> 4) << 1;  // lanes 16-31 hold K+2,K+3

    const float* Ws = W + (size_t)s * DIMV * DIMV;
    int col[4];
    #pragma unroll
    for (int j = 0; j < 4; ++j) col[j] = wave * 64 + j * 16 + row16;

    v8f acc[4] = {};

    for (int k0 = 0; k0 < DIMV; k0 += 4) {
        // A fragment: lane -> mean[row16][k0+khalf], [k0+khalf+1]
        const float* ap = &meanLds[row16 * LDSROW + k0 + khalf];
        v2f a;
        a.x = ap[0];
        a.y = ap[1];
        #pragma unroll
        for (int j = 0; j < 4; ++j) {
            // B fragment: lane -> W[k0+khalf][col], W[k0+khalf+1][col]
            const float* bp = Ws + (size_t)(k0 + khalf) * DIMV + col[j];
            v2f bb;
            bb.x = bp[0];
            bb.y = bp[DIMV];
            acc[j] = __builtin_amdgcn_wmma_f32_16x16x4_f32(
                false, a, false, bb, (short)0, acc[j], false, false);
        }
    }

    // ---- epilogue: bias + tanh, scatter rows to enc ----
    #pragma unroll
    for (int j = 0; j < 4; ++j) {
        float bias = bvec[s * DIMV + col[j]];
        #pragma unroll
        for (int r = 0; r < 8; ++r) {
            int rrow = r + ((lane >> 4) ? 8 : 0);   // C layout: VGPR r -> M=r / M=r+8
            int node = nodeIds[rrow];
            if (node >= 0) {
                float val = tanhf(acc[j][r] + bias);
                enc[(size_t)node * DIMV + col[j]] = val;
            }
        }
    }
}

// ---------------------------------------------------------------------------
// out = enc[0] @ W_out + b_out   (256 -> 10)
// ---------------------------------------------------------------------------
__global__ __launch_bounds__(256) void rnc_out(const float* __restrict__ enc,
                                               const float* __restrict__ W_out,
                                               const float* __restrict__ b_out,
                                               float* __restrict__ out) {
    __shared__ float red[256];
    int tid = threadIdx.x;
    float e = enc[tid];                  // enc[root][tid]
    for (int o = 0; o < OUTV; ++o) {
        red[tid] = e * W_out[tid * OUTV + o];
        __syncthreads();
        for (int st = 128; st > 0; st >>= 1) {
            if (tid < st) red[tid] += red[tid + st];
            __syncthreads();
        }
        if (tid == 0) out[o] = red[0] + b_out[o];
        __syncthreads();
    }
}

// ---------------------------------------------------------------------------
extern "C" void kernel_launch(void* const* d_in, const int* in_sizes, int n_in,
                              void* d_out, int out_size, void* d_ws, size_t ws_size,
                              hipStream_t stream) {
    const int*   x_sym     = (const int*)d_in[0];
    const int*   child_idx = (const int*)d_in[1];
    const float* W         = (const float*)d_in[2];
    const float* b         = (const float*)d_in[3];
    const float* W_out     = (const float*)d_in[4];
    const float* b_out     = (const float*)d_in[5];
    float*       out       = (float*)d_out;

    const int N = in_sizes[0];                  // 16383
    int depth = 0;
    while (((1 << depth) - 1) < N) ++depth;     // 14

    char* ws = (char*)d_ws;
    float* enc    = (float*)ws;                                        // N*256 f32
    int*   lists  = (int*)(ws + (size_t)N * DIMV * sizeof(float));     // S*N int
    int*   counts = (int*)(ws + (size_t)N * DIMV * sizeof(float)
                              + (size_t)SV * N * sizeof(int));         // depth*S int

    rnc_zero_counts<<<1, 256, 0, stream>>>(counts, depth * SV);
    rnc_bucket<<<(N + 255) / 256, 256, 0, stream>>>(x_sym, lists, counts, N);

    for (int d = depth - 1; d >= 0; --d) {
        int lo  = (1 << d) - 1;
        int hi  = (2 << d) - 1;
        if (hi > N) hi = N;
        int lsz = hi - lo;
        dim3 grid(SV, (lsz + 15) / 16);
        rnc_level_gemm<<<grid, 128, 0, stream>>>(child_idx, W, b, lists, counts,
                                                 enc, d, lo, N);
    }

    rnc_out<<<1, 256, 0, stream>>>(enc, W_out, b_out, out);
    (void)n_in; (void)out_size; (void)ws_size;
}